// NerfModule_83811991814328
// MI455X (gfx1250) — compile-verified
//
#include <hip/hip_runtime.h>
#include <math.h>

// ---------------------------------------------------------------------------
// NeRF sample_along_rays + positional encoding for MI455X (gfx1250, wave32).
//
// Roofline: output = 8192*128*120 f32 = 503 MB  ->  ~21.6 us @ 23.3 TB/s.
// Memory-bound; transcendental work (63M v_sin/v_cos) stays under that floor
// because the per-ray view encoding is computed once, not per sample.
//
// CDNA5-specific paths:
//   * global_load_async_to_lds_b32 + s_wait_asynccnt (ASYNCcnt-tracked DMA)
//   * nontemporal (TH=NT) streaming 128-bit stores for the 503 MB output
//   * v_fract/v_sin/v_cos native encode (exact power-of-2 octave doubling)
//   * wave32 block layout, LDS-staged fully-coalesced output stream
// ---------------------------------------------------------------------------

#define N_BINS   128
#define S_SAMP   128
#define DEG      10
#define EPS_V    1e-5f
#define INV_2PI  0.15915493667125702f   // 1/(2*pi), fp32

typedef __attribute__((ext_vector_type(4))) float f32x4;   // native vector for NT stores

__device__ __forceinline__ void async_f32_to_lds(const float* g, float* l) {
    // GV addressing: per-lane 64-bit global addr, per-lane LDS byte addr.
    unsigned lds = (unsigned)(unsigned long long)(void*)l;  // low 32b of shared
    asm volatile("global_load_async_to_lds_b32 %0, %1, off"
                 :: "v"(lds), "v"(g) : "memory");
}

extern "C" __global__ void __launch_bounds__(128)
nerf_sample_encode(const float* __restrict__ origins,
                   const float* __restrict__ directions,
                   const float* __restrict__ bins,
                   const float* __restrict__ weights,
                   const float* __restrict__ u_rand,
                   float* __restrict__ out)
{
    __shared__ float sW[N_BINS];
    __shared__ float sBins[N_BINS];
    __shared__ float sScanA[N_BINS];
    __shared__ float sScanB[N_BINS];
    __shared__ float sCdf[N_BINS];
    __shared__ __align__(16) float sView[64];            // 60 used
    __shared__ __align__(16) float sPos[S_SAMP * 60];    // 30 KB staging tile

    const int tid = threadIdx.x;   // == sample index s
    const int ray = blockIdx.x;

    // ---- 1) stage weights & bins into LDS via CDNA5 async DMA -------------
    async_f32_to_lds(weights + (size_t)ray * N_BINS + tid, &sW[tid]);
    async_f32_to_lds(bins    + (size_t)ray * N_BINS + tid, &sBins[tid]);

    // per-ray origin / direction (uniform -> scalar loads)
    const float ox = origins[3 * ray + 0];
    const float oy = origins[3 * ray + 1];
    const float oz = origins[3 * ray + 2];
    const float dx = directions[3 * ray + 0];
    const float dy = directions[3 * ray + 1];
    const float dz = directions[3 * ray + 2];

    // view encoding: constant per ray, compute once (threads 0..29)
    if (tid < 30) {
        const int d = tid / 3, c = tid - 3 * d;
        const float comp = (c == 0) ? dx : ((c == 1) ? dy : dz);
        // revolutions; octave scale 2^d is exact in fp32
        float rv = __builtin_amdgcn_fractf((comp * INV_2PI) * (float)(1 << d));
        sView[tid]      = __builtin_amdgcn_sinf(rv);   // sin(x*2^d)
        sView[30 + tid] = __builtin_amdgcn_cosf(rv);   // sin(x*2^d + pi/2)
    }

    // my stratified u (read-once -> nontemporal load)
    const float ur = __builtin_nontemporal_load(u_rand + (size_t)ray * S_SAMP + tid);
    const float uval = ((float)tid + ur) * (1.0f / (float)S_SAMP);

    asm volatile("s_wait_asynccnt 0x0" ::: "memory");
    __syncthreads();

    // ---- 2) inclusive scan of w[0..126] (slot 127 = 0), ping-pong buffers --
    sScanA[tid] = (tid < N_BINS - 1) ? sW[tid] : 0.0f;
    __syncthreads();
    float* src = sScanA;
    float* dst = sScanB;
#pragma unroll
    for (int off = 1; off < 128; off <<= 1) {
        float v = src[tid];
        if (tid >= off) v += src[tid - off];
        dst[tid] = v;
        __syncthreads();
        float* t = src; src = dst; dst = t;
    }
    // inclusive prefix sums now in src[]

    // cdf[0]=0; cdf[1+j]=min(1,(cs[j]+(j+1)*pad/127)/wsum'), j=0..125; cdf[127]=1
    {
        const float wsum   = src[126];
        const float pad    = fmaxf(0.0f, EPS_V - wsum);
        const float wsumP  = wsum + pad;
        const float padPer = pad * (1.0f / 127.0f);
        if (tid == 0)   sCdf[0] = 0.0f;
        if (tid <= 125) sCdf[tid + 1] =
            fminf(1.0f, (src[tid] + (float)(tid + 1) * padPer) / wsumP);
        if (tid == 127) sCdf[127] = 1.0f;
    }
    __syncthreads();

    // ---- 3) searchsorted(cdf, u, 'right') == upper_bound ------------------
    int inds = 0;
#pragma unroll
    for (int step = 64; step >= 1; step >>= 1) {
        const int cand = inds + step;
        if (cand <= 128 && sCdf[cand - 1] <= uval) inds = cand;
    }
    int below = inds - 1; below = (below < 0) ? 0 : ((below > 127) ? 127 : below);
    int above = (inds > 127) ? 127 : inds;

    const float c0 = sCdf[below], c1 = sCdf[above];
    const float b0 = sBins[below], b1 = sBins[above];
    float denom = c1 - c0;
    if (denom < EPS_V) denom = 1.0f;
    const float t  = (uval - c0) / denom;
    const float sv = b0 + t * (b1 - b0);

    const float px = ox + sv * dx;
    const float py = oy + sv * dy;
    const float pz = oz + sv * dz;

    // ---- 4) positional encoding of the point into the LDS tile ------------
    // layout per sample: [ sin d0(x,y,z) .. sin d9 | cos d0 .. cos d9 ]
    {
        float rvx = __builtin_amdgcn_fractf(px * INV_2PI);
        float rvy = __builtin_amdgcn_fractf(py * INV_2PI);
        float rvz = __builtin_amdgcn_fractf(pz * INV_2PI);
        float* row = sPos + tid * 60;
#pragma unroll
        for (int d = 0; d < DEG; ++d) {
            row[d * 3 + 0]      = __builtin_amdgcn_sinf(rvx);
            row[30 + d * 3 + 0] = __builtin_amdgcn_cosf(rvx);
            row[d * 3 + 1]      = __builtin_amdgcn_sinf(rvy);
            row[30 + d * 3 + 1] = __builtin_amdgcn_cosf(rvy);
            row[d * 3 + 2]      = __builtin_amdgcn_sinf(rvz);
            row[30 + d * 3 + 2] = __builtin_amdgcn_cosf(rvz);
            rvx = __builtin_amdgcn_fractf(rvx + rvx);   // frac(2x): exact octave step
            rvy = __builtin_amdgcn_fractf(rvy + rvy);
            rvz = __builtin_amdgcn_fractf(rvz + rvz);
        }
    }
    __syncthreads();

    // ---- 5) coalesced nontemporal 128-bit stream: [pos(60)|view(60)] ------
    f32x4* out4 = (f32x4*)out;
    const size_t base4 = (size_t)ray * (S_SAMP * 120 / 4);   // 3840 vec4/ray
    const f32x4* view4 = (const f32x4*)sView;
#pragma unroll
    for (int it = 0; it < 30; ++it) {
        const int q = it * 128 + tid;          // vec4 index within ray
        const int f = q * 4;                   // float index
        const int s = f / 120;                 // sample
        const int k = f - s * 120;             // channel (multiple of 4)
        f32x4 v;
        if (k < 60) v = *(const f32x4*)(sPos + s * 60 + k);
        else        v = view4[(k - 60) >> 2];
        __builtin_nontemporal_store(v, &out4[base4 + q]);   // TH=NT streaming
    }
}

extern "C" void kernel_launch(void* const* d_in, const int* in_sizes, int n_in,
                              void* d_out, int out_size, void* d_ws, size_t ws_size,
                              hipStream_t stream) {
    (void)n_in; (void)out_size; (void)d_ws; (void)ws_size;
    const float* origins    = (const float*)d_in[0];
    const float* directions = (const float*)d_in[1];
    const float* bins       = (const float*)d_in[2];
    const float* weights    = (const float*)d_in[3];
    const float* u_rand     = (const float*)d_in[4];
    float* out = (float*)d_out;

    const int R = in_sizes[0] / 3;   // 8192
    hipLaunchKernelGGL(nerf_sample_encode, dim3(R), dim3(S_SAMP), 0, stream,
                       origins, directions, bins, weights, u_rand, out);
}